// GCN_17600775979431
// MI455X (gfx1250) — compile-verified
//
#include <hip/hip_runtime.h>
#include <hip/hip_bf16.h>
#include <math.h>

typedef __attribute__((ext_vector_type(2))) float v2f;
typedef __attribute__((ext_vector_type(8))) float v8f;

// ---------------------------------------------------------------------------
// Degree: deg[d] += 1 per edge (self-loop +1 folded into dinv kernel)
// ---------------------------------------------------------------------------
__global__ __launch_bounds__(256) void deg_kernel(const int* __restrict__ dst,
                                                  float* __restrict__ deg, int E) {
    int e = blockIdx.x * blockDim.x + threadIdx.x;
    if (e < E) atomicAdd(&deg[dst[e]], 1.0f);
}

__global__ __launch_bounds__(256) void dinv_kernel(const float* __restrict__ deg,
                                                   float* __restrict__ dinv, int N) {
    int i = blockIdx.x * blockDim.x + threadIdx.x;
    if (i < N) dinv[i] = rsqrtf(deg[i] + 1.0f);
}

// ---------------------------------------------------------------------------
// C[M,NN] = A[M,K] @ B[K,NN], fp32, one wave per 16x16 tile using
// V_WMMA_F32_16X16X4_F32.  K, NN are compile-time so the main loop is
// completely unguarded; B is staged into LDS zero-padded to KP x NNP, which
// makes the K/N tails free (zero B rows/cols annihilate any A value, so A
// only needs an in-bounds index clamp on the single tail iteration).
// Requires M % 16 == 0 (true here: 100000 = 6250*16).
// ---------------------------------------------------------------------------
template <int K, int NN>
__global__ __launch_bounds__(256) void gemm_wmma(const float* __restrict__ A,
                                                 const float* __restrict__ B,
                                                 float* __restrict__ C, int M) {
    constexpr int NTN = (NN + 15) / 16;   // 16-wide col tiles
    constexpr int NNP = NTN * 16;         // padded N
    constexpr int KP  = (K + 3) & ~3;     // padded K
    __shared__ float Bs[KP * NNP];

    // Cooperative zero-padded stage of the (tiny) weight matrix.
    for (int idx = threadIdx.x; idx < KP * NNP; idx += 256) {
        const int k = idx / NNP, n = idx % NNP;
        Bs[idx] = (k < K && n < NN) ? B[k * NN + n] : 0.0f;
    }
    __syncthreads();

    const int wave = blockIdx.x * 8 + (threadIdx.x >> 5);
    const int lane = threadIdx.x & 31;
    const int mt = wave / NTN;
    const int nt = wave % NTN;
    if (mt * 16 >= M) return;             // wave-uniform: EXEC stays full for WMMA

    const int mrow = mt * 16 + (lane & 15);       // A row fed by this lane
    const int ncol = nt * 16 + (lane & 15);       // padded B column fed by this lane
    const int koff = (lane >> 4) << 1;            // lanes 0-15: k0,k0+1  16-31: k0+2,k0+3
    const float* __restrict__ arow = A + (size_t)mrow * K;

    v8f acc = {};
    int k0 = 0;
    // Main loop: all indices provably in range -> unconditional loads, no EXEC games.
    for (; k0 + 4 <= K; k0 += 4) {
        const int ka = k0 + koff;
        v2f a, b;
        a.x = arow[ka];
        a.y = arow[ka + 1];
        b.x = Bs[ka * NNP + ncol];
        b.y = Bs[(ka + 1) * NNP + ncol];
        acc = __builtin_amdgcn_wmma_f32_16x16x4_f32(
            false, a, false, b, (short)0, acc, false, false);
    }
    // Tail (only exists for K % 4 != 0, i.e. layer 1's K=165): clamp A index,
    // correctness guaranteed by zero rows in Bs.
    if constexpr ((K & 3) != 0) {
        const int ka = k0 + koff;
        const int c0 = (ka     < K) ? ka     : (K - 1);
        const int c1 = (ka + 1 < K) ? (ka + 1) : (K - 1);
        v2f a, b;
        a.x = arow[c0];
        a.y = arow[c1];
        b.x = Bs[ka * NNP + ncol];         // ka, ka+1 <= KP-1: in-bounds, zero rows
        b.y = Bs[(ka + 1) * NNP + ncol];
        acc = __builtin_amdgcn_wmma_f32_16x16x4_f32(
            false, a, false, b, (short)0, acc, false, false);
    }

    // D layout: lanes 0-15 hold M = mt*16 + r (VGPR r=0..7), lanes 16-31 hold M+8.
    const int half = lane >> 4;
    const int ncl = nt * 16 + (lane & 15);
    if (NN >= NNP || (lane & 15) < NN) {   // compile-time true except NN=2 layer
        const int mbase = mt * 16 + half * 8;
#pragma unroll
        for (int r = 0; r < 8; ++r)
            C[(size_t)(mbase + r) * NN + ncl] = acc[r];
    }
}

// ---------------------------------------------------------------------------
// Edge aggregation: agg[dst] += h[src] * dinv[src]*dinv[dst]
// F is a power-of-two multiple of 4: one thread per (edge, float4 chunk).
// ---------------------------------------------------------------------------
template <int F>
__global__ __launch_bounds__(256) void edge_agg4(const int* __restrict__ src,
                                                 const int* __restrict__ dst,
                                                 const float* __restrict__ dinv,
                                                 const float* __restrict__ h,
                                                 float* __restrict__ agg, int E) {
    constexpr int CPE = F >> 2;                       // chunks per edge (8 or 4)
    const long long tid = (long long)blockIdx.x * 256 + threadIdx.x;
    if (tid >= (long long)E * CPE) return;
    const int e = (int)(tid / CPE);
    const int c = ((int)tid & (CPE - 1)) << 2;
    const int s = src[e], d = dst[e];
    const float nrm = dinv[s] * dinv[d];
    const float4 v = *(const float4*)(h + (size_t)s * F + c);
    float* ap = agg + (size_t)d * F + c;
    atomicAdd(ap + 0, v.x * nrm);
    atomicAdd(ap + 1, v.y * nrm);
    atomicAdd(ap + 2, v.z * nrm);
    atomicAdd(ap + 3, v.w * nrm);
}

// F == 2 path: one thread per edge.
__global__ __launch_bounds__(256) void edge_agg2(const int* __restrict__ src,
                                                 const int* __restrict__ dst,
                                                 const float* __restrict__ dinv,
                                                 const float* __restrict__ h,
                                                 float* __restrict__ agg, int E) {
    const int e = blockIdx.x * 256 + threadIdx.x;
    if (e >= E) return;
    const int s = src[e], d = dst[e];
    const float nrm = dinv[s] * dinv[d];
    const float2 v = *(const float2*)(h + (size_t)s * 2);
    float* ap = agg + (size_t)d * 2;
    atomicAdd(ap + 0, v.x * nrm);
    atomicAdd(ap + 1, v.y * nrm);
}

// ---------------------------------------------------------------------------
// Epilogue: out = [relu](agg + h * dinv^2 + bias)   (in-place into agg OK)
// ---------------------------------------------------------------------------
template <int F, int RELU>
__global__ __launch_bounds__(256) void epilogue_kernel(const float* __restrict__ agg,
                                                       const float* __restrict__ h,
                                                       const float* __restrict__ dinv,
                                                       const float* __restrict__ bias,
                                                       float* __restrict__ out, int N) {
    const long long tid = (long long)blockIdx.x * 256 + threadIdx.x;
    if (tid >= (long long)N * F) return;
    const int i = (int)(tid >> __builtin_ctz(F));
    const int f = (int)tid & (F - 1);
    const float di = dinv[i];
    float v = agg[tid] + h[tid] * di * di + bias[f];
    if (RELU) v = fmaxf(v, 0.0f);
    out[tid] = v;
}

// ---------------------------------------------------------------------------
// log_softmax over 2 classes
// ---------------------------------------------------------------------------
__global__ __launch_bounds__(256) void logsoftmax2_kernel(const float* __restrict__ h,
                                                          float* __restrict__ out, int N) {
    const int i = blockIdx.x * 256 + threadIdx.x;
    if (i >= N) return;
    const float a = h[2 * i], b = h[2 * i + 1];
    const float m = fmaxf(a, b);
    const float lse = m + logf(expf(a - m) + expf(b - m));
    out[2 * i]     = a - lse;
    out[2 * i + 1] = b - lse;
}

// ---------------------------------------------------------------------------
// Launch
// ---------------------------------------------------------------------------
static inline int cdiv(long long a, long long b) { return (int)((a + b - 1) / b); }

extern "C" void kernel_launch(void* const* d_in, const int* in_sizes, int n_in,
                              void* d_out, int out_size, void* d_ws, size_t ws_size,
                              hipStream_t stream) {
    const float* x  = (const float*)d_in[0];
    const int*   ei = (const int*)d_in[1];
    const float* W1 = (const float*)d_in[2];
    const float* b1 = (const float*)d_in[3];
    const float* W2 = (const float*)d_in[4];
    const float* b2 = (const float*)d_in[5];
    const float* W3 = (const float*)d_in[6];
    const float* b3 = (const float*)d_in[7];

    const int D = 165;
    const int N = in_sizes[0] / D;     // 100000 (multiple of 16)
    const int E = in_sizes[1] / 2;     // 3200000
    const int* src = ei;
    const int* dst = ei + E;

    // workspace (floats); zeroed prefix = deg + agg1 + agg2 + agg3
    float* ws   = (float*)d_ws;
    float* deg  = ws;
    float* agg1 = deg  + (size_t)N;
    float* agg2 = agg1 + (size_t)N * 32;
    float* agg3 = agg2 + (size_t)N * 16;
    float* dinv = agg3 + (size_t)N * 2;
    float* h1   = dinv + (size_t)N;
    float* h2   = h1   + (size_t)N * 32;
    float* h3   = h2   + (size_t)N * 16;
    const size_t zero_bytes = (size_t)N * (1 + 32 + 16 + 2) * sizeof(float);

    hipMemsetAsync(d_ws, 0, zero_bytes, stream);

    deg_kernel<<<cdiv(E, 256), 256, 0, stream>>>(dst, deg, E);
    dinv_kernel<<<cdiv(N, 256), 256, 0, stream>>>(deg, dinv, N);

    const int mt = N / 16;

    // ---- layer 1: 165 -> 32, relu ----
    gemm_wmma<165, 32><<<cdiv((long long)mt * 2, 8), 256, 0, stream>>>(x, W1, h1, N);
    edge_agg4<32><<<cdiv((long long)E * 8, 256), 256, 0, stream>>>(src, dst, dinv, h1, agg1, E);
    epilogue_kernel<32, 1><<<cdiv((long long)N * 32, 256), 256, 0, stream>>>(agg1, h1, dinv, b1, agg1, N);

    // ---- layer 2: 32 -> 16, relu ----
    gemm_wmma<32, 16><<<cdiv((long long)mt, 8), 256, 0, stream>>>(agg1, W2, h2, N);
    edge_agg4<16><<<cdiv((long long)E * 4, 256), 256, 0, stream>>>(src, dst, dinv, h2, agg2, E);
    epilogue_kernel<16, 1><<<cdiv((long long)N * 16, 256), 256, 0, stream>>>(agg2, h2, dinv, b2, agg2, N);

    // ---- layer 3: 16 -> 2, then log_softmax ----
    gemm_wmma<16, 2><<<cdiv((long long)mt, 8), 256, 0, stream>>>(agg2, W3, h3, N);
    edge_agg2<<<cdiv(E, 256), 256, 0, stream>>>(src, dst, dinv, h3, agg3, E);
    epilogue_kernel<2, 0><<<cdiv((long long)N * 2, 256), 256, 0, stream>>>(agg3, h3, dinv, b3, agg3, N);
    logsoftmax2_kernel<<<cdiv(N, 256), 256, 0, stream>>>(agg3, (float*)d_out, N);
}